// STGS_9199819948604
// MI455X (gfx1250) — compile-verified
//
#include <hip/hip_runtime.h>
#include <stdint.h>

// ---- problem constants (2, 256, 50257) fp32 ----
#define VOCAB     50257
#define RPAIRS    256            // blocks; block r handles rows r and r+256
#define HALF_N    12865792u      // 256*50257 == total/2 (batch boundary)
#define THREADS   512
#define NFULL     98             // 98*512 = 50176 fully-covered columns
#define TAILBASE  50176
// output layout (flat floats): ids[512] | one_hot[512*V] | temp[1] | y_soft[512*V]
#define OFF_OH    512
#define OFF_TEMP  25732096       // 512 + 512*50257
#define OFF_YS    25732097

// ---- JAX threefry2x32 (20 rounds), usable host+device ----
__host__ __device__ inline void threefry2x32(uint32_t k0, uint32_t k1,
                                             uint32_t c0, uint32_t c1,
                                             uint32_t& o0, uint32_t& o1) {
  const uint32_t k2 = k0 ^ k1 ^ 0x1BD11BDAu;
  uint32_t x0 = c0 + k0;
  uint32_t x1 = c1 + k1;
#define TF4(r0, r1, r2, r3)                                                  \
  x0 += x1; x1 = (x1 << r0) | (x1 >> (32 - r0)); x1 ^= x0;                   \
  x0 += x1; x1 = (x1 << r1) | (x1 >> (32 - r1)); x1 ^= x0;                   \
  x0 += x1; x1 = (x1 << r2) | (x1 >> (32 - r2)); x1 ^= x0;                   \
  x0 += x1; x1 = (x1 << r3) | (x1 >> (32 - r3)); x1 ^= x0;
  TF4(13, 15, 26, 6)  x0 += k1; x1 += k2 + 1u;
  TF4(17, 29, 16, 24) x0 += k2; x1 += k0 + 2u;
  TF4(13, 15, 26, 6)  x0 += k0; x1 += k1 + 3u;
  TF4(17, 29, 16, 24) x0 += k1; x1 += k2 + 4u;
  TF4(13, 15, 26, 6)  x0 += k2; x1 += k0 + 5u;
#undef TF4
  o0 = x0;
  o1 = x1;
}

__device__ inline float bits_to_u01(uint32_t b) {
  return __uint_as_float((b >> 9) | 0x3f800000u) - 1.0f;  // [0,1)
}

__global__ __launch_bounds__(THREADS) void stgs_kernel(
    const float* __restrict__ x, float* __restrict__ out,
    uint32_t kg0, uint32_t kg1, uint32_t ks0, uint32_t ks1) {
  const int rp   = blockIdx.x;          // row pair: rows rp (batch0), rp+256 (batch1)
  const int tid  = threadIdx.x;
  const int lane = tid & 31;
  const int wave = tid >> 5;
  const uint32_t rb0 = (uint32_t)rp * (uint32_t)VOCAB;   // < HALF_N always
  const uint32_t rb1 = rb0 + HALF_N;

  float* outIds = out;
  float* oh0 = out + OFF_OH + (size_t)rb0;
  float* oh1 = out + OFF_OH + (size_t)rb1;
  float* tr0 = out + OFF_YS + (size_t)rb0;  // y_soft slot doubles as t-scratch
  float* tr1 = out + OFF_YS + (size_t)rb1;
  const float* x0r = x + (size_t)rb0;
  const float* x1r = x + (size_t)rb1;

  __shared__ float stage[2][2][THREADS];    // [buf][row-of-pair][tid]
  __shared__ float red_m[2][16], red_s[2][16], red_z[2][16];
  __shared__ int   red_i[2][16];
  __shared__ float bc_m[2], bc_invs[2];
  __shared__ int   bc_idx[2];

  // per-row online-softmax + gumbel-max argmax state
  float m0 = -3.4e38f, s0 = 0.0f, bz0 = -3.4e38f;
  float m1 = -3.4e38f, s1 = 0.0f, bz1 = -3.4e38f;
  int bi0 = 0x7fffffff, bi1 = 0x7fffffff;

  // One threefry call per stream feeds BOTH batches (word0->batch0, word1->batch1).
  auto process = [&](int col, float xv0, float xv1) {
    const uint32_t c0 = rb0 + (uint32_t)col;   // counter pair (c0, c0+HALF_N)
    const uint32_t c1 = c0 + HALF_N;
    uint32_t ga, gb;
    threefry2x32(kg0, kg1, c0, c1, ga, gb);
    const float u0 = bits_to_u01(ga) * (0.999f - 1e-12f) + 1e-12f;
    const float u1 = bits_to_u01(gb) * (0.999f - 1e-12f) + 1e-12f;
    const float t0 = xv0 + (-__logf(-__logf(u0)));
    const float t1 = xv1 + (-__logf(-__logf(u1)));
    // online softmax, 1 exp per element
    if (t0 > m0) { s0 = s0 * __expf(m0 - t0) + 1.0f; m0 = t0; }
    else         { s0 += __expf(t0 - m0); }
    if (t1 > m1) { s1 = s1 * __expf(m1 - t1) + 1.0f; m1 = t1; }
    else         { s1 += __expf(t1 - m1); }
    // categorical sampling noise (k_sample stream), gumbel-max
    uint32_t sa, sb;
    threefry2x32(ks0, ks1, c0, c1, sa, sb);
    const float ub0 = bits_to_u01(sa) + 1.17549435e-38f;
    const float ub1 = bits_to_u01(sb) + 1.17549435e-38f;
    const float z0 = t0 + (-__logf(-__logf(ub0)));
    const float z1 = t1 + (-__logf(-__logf(ub1)));
    if (z0 > bz0 || (z0 == bz0 && col < bi0)) { bz0 = z0; bi0 = col; }
    if (z1 > bz1 || (z1 == bz1 && col < bi1)) { bz1 = z1; bi1 = col; }
    // stash t (RT store; re-read in pass B from L2)
    tr0[col] = t0;
    tr1[col] = t1;
  };

  // ---- pass A: double-buffered async-to-LDS streaming of both x rows ----
  {
    uint32_t la = (uint32_t)(uintptr_t)&stage[0][0][tid];
    uint32_t lb = (uint32_t)(uintptr_t)&stage[0][1][tid];
    const float* ga = x0r + tid;
    const float* gb = x1r + tid;
    asm volatile("global_load_async_to_lds_b32 %0, %1, off" :: "v"(la), "v"(ga) : "memory");
    asm volatile("global_load_async_to_lds_b32 %0, %1, off" :: "v"(lb), "v"(gb) : "memory");
  }
  for (int it = 0; it < NFULL; ++it) {
    if (it + 1 < NFULL) {
      const int nb = (it + 1) & 1;
      uint32_t la = (uint32_t)(uintptr_t)&stage[nb][0][tid];
      uint32_t lb = (uint32_t)(uintptr_t)&stage[nb][1][tid];
      const float* ga = x0r + (it + 1) * THREADS + tid;
      const float* gb = x1r + (it + 1) * THREADS + tid;
      asm volatile("global_load_async_to_lds_b32 %0, %1, off" :: "v"(la), "v"(ga) : "memory");
      asm volatile("global_load_async_to_lds_b32 %0, %1, off" :: "v"(lb), "v"(gb) : "memory");
      asm volatile("s_wait_asynccnt 2" ::: "memory");
    } else {
      asm volatile("s_wait_asynccnt 0" ::: "memory");
    }
    process(it * THREADS + tid, stage[it & 1][0][tid], stage[it & 1][1][tid]);
  }
  {  // tail (81 columns), plain loads
    const int col = TAILBASE + tid;
    if (col < VOCAB) process(col, x0r[col], x1r[col]);
  }

  // ---- block reduction: wave32 shuffles, then LDS across 16 waves ----
  for (int o = 16; o > 0; o >>= 1) {
    float om = __shfl_down(m0, o, 32), os = __shfl_down(s0, o, 32);
    float mn = fmaxf(m0, om);
    s0 = s0 * __expf(m0 - mn) + os * __expf(om - mn);  m0 = mn;
    om = __shfl_down(m1, o, 32); os = __shfl_down(s1, o, 32);
    mn = fmaxf(m1, om);
    s1 = s1 * __expf(m1 - mn) + os * __expf(om - mn);  m1 = mn;
    float oz = __shfl_down(bz0, o, 32); int oi = __shfl_down(bi0, o, 32);
    if (oz > bz0 || (oz == bz0 && oi < bi0)) { bz0 = oz; bi0 = oi; }
    oz = __shfl_down(bz1, o, 32); oi = __shfl_down(bi1, o, 32);
    if (oz > bz1 || (oz == bz1 && oi < bi1)) { bz1 = oz; bi1 = oi; }
  }
  if (lane == 0) {
    red_m[0][wave] = m0; red_s[0][wave] = s0; red_z[0][wave] = bz0; red_i[0][wave] = bi0;
    red_m[1][wave] = m1; red_s[1][wave] = s1; red_z[1][wave] = bz1; red_i[1][wave] = bi1;
  }
  __syncthreads();
  if (tid < 2) {   // thread 0 -> row pair member 0, thread 1 -> member 1
    const int r = tid;
    float M = red_m[r][0], S = red_s[r][0], Z = red_z[r][0];
    int   I = red_i[r][0];
    for (int w = 1; w < 16; ++w) {
      const float mw = red_m[r][w];
      const float mn = fmaxf(M, mw);
      S = S * __expf(M - mn) + red_s[r][w] * __expf(mw - mn);
      M = mn;
      if (red_z[r][w] > Z || (red_z[r][w] == Z && red_i[r][w] < I)) {
        Z = red_z[r][w]; I = red_i[r][w];
      }
    }
    bc_m[r] = M; bc_invs[r] = 1.0f / S; bc_idx[r] = I;
    outIds[rp + r * 256] = (float)I;            // diff_output_ids forward value
    if (rp == 0 && r == 0) out[OFF_TEMP] = 1.0f;  // temp
  }
  __syncthreads();
  const float M0 = bc_m[0], iS0 = bc_invs[0];
  const float M1 = bc_m[1], iS1 = bc_invs[1];
  const int   I0 = bc_idx[0], I1 = bc_idx[1];

  // ---- pass B: finalize y_soft + one_hot for both rows (streaming NT stores) ----
  for (int col = tid; col < VOCAB; col += THREADS) {
    const float t0 = tr0[col];   // written by this same thread; L2-resident
    const float t1 = tr1[col];
    __builtin_nontemporal_store(__expf(t0 - M0) * iS0, &tr0[col]);
    __builtin_nontemporal_store(__expf(t1 - M1) * iS1, &tr1[col]);
    __builtin_nontemporal_store((col == I0) ? 1.0f : 0.0f, &oh0[col]);
    __builtin_nontemporal_store((col == I1) ? 1.0f : 0.0f, &oh1[col]);
  }
}

extern "C" void kernel_launch(void* const* d_in, const int* in_sizes, int n_in,
                              void* d_out, int out_size, void* d_ws, size_t ws_size,
                              hipStream_t stream) {
  (void)in_sizes; (void)n_in; (void)out_size; (void)d_ws; (void)ws_size;
  const float* x = (const float*)d_in[0];
  float* out = (float*)d_out;

  // Reproduce jax.random.split(jax.random.key(42)) on host (deterministic):
  // counts [0,1,2,3] -> pairs (0,2),(1,3); out=[a0,b0,a1,b1]; keys=reshape(2,2)
  uint32_t a0, a1, b0, b1;
  threefry2x32(0u, 42u, 0u, 2u, a0, a1);
  threefry2x32(0u, 42u, 1u, 3u, b0, b1);
  // k_gumbel = (a0, b0), k_sample = (a1, b1)
  stgs_kernel<<<RPAIRS, THREADS, 0, stream>>>(x, out, a0, b0, a1, b1);
}